// CoAttNet_85263690760310
// MI455X (gfx1250) — compile-verified
//
#include <hip/hip_runtime.h>
#include <hip/hip_bf16.h>

typedef __attribute__((ext_vector_type(2))) float v2f;
typedef __attribute__((ext_vector_type(8))) float v8f;

#define BM 64
#define BN 64
#define BK 16

// ---------------------------------------------------------------------------
// gfx1250 async global->LDS copy (ASYNCcnt-tracked), per CDNA5 ISA.
// lds_addr: LDS byte address (low 32 bits of generic pointer to __shared__).
// ---------------------------------------------------------------------------
__device__ __forceinline__ void async_g2l_b32(unsigned lds_addr, const float* g)
{
    asm volatile("global_load_async_to_lds_b32 %0, %1, off"
                 :: "v"(lds_addr), "v"((unsigned long long)g)
                 : "memory");
}
__device__ __forceinline__ void async_wait_all()
{
    asm volatile("s_wait_asynccnt 0x0" ::: "memory");
}
__device__ __forceinline__ unsigned lds_addr_of(const void* p)
{
    return (unsigned)(unsigned long long)p;
}

// ---------------------------------------------------------------------------
// Generic batched strided GEMM: C[b][i][j] = sum_k A(i,k) * B(k,j)
// A element (i,k) at Ab + b*Abat + i*Ars + k*Acs
// B element (k,j) at Bb + b*Bbat + k*Brs + j*Bcs
// C row-major [M x Nn], ldc = Nn.
// V_WMMA_F32_16X16X4_F32; 128 threads = 4 waves, each wave 32x32 of a 64x64
// tile; full tiles staged with GLOBAL_LOAD_ASYNC_TO_LDS_B32.
// ---------------------------------------------------------------------------
__global__ __launch_bounds__(128)
void gemm_f32_wmma(const float* __restrict__ Ab, const float* __restrict__ Bb,
                   float* __restrict__ Cb, int M, int Nn, int K,
                   long Ars, long Acs, long Brs, long Bcs,
                   long Abat, long Bbat, long Cbat)
{
    __shared__ float As[BM][BK + 1];
    __shared__ float Bs[BK][BN + 1];

    const int nb = blockIdx.z;
    const float* Ap = Ab + (long)nb * Abat;
    const float* Bp = Bb + (long)nb * Bbat;
    float* Cp = Cb + (long)nb * Cbat;

    const int m0 = blockIdx.y * BM;
    const int n0 = blockIdx.x * BN;
    const int tid = threadIdx.x;
    const int wave = tid >> 5;
    const int lane = tid & 31;
    const int wr = (wave >> 1) * 32;   // wave row offset inside tile
    const int wc = (wave & 1) * 32;    // wave col offset inside tile
    const int half = lane >> 4;        // K-half selector (ISA lane split)
    const int lr = lane & 15;

    v8f acc[2][2];
#pragma unroll
    for (int i = 0; i < 2; ++i)
#pragma unroll
        for (int j = 0; j < 2; ++j)
            acc[i][j] = (v8f){0.f,0.f,0.f,0.f,0.f,0.f,0.f,0.f};

    const bool mn_full = (m0 + BM <= M) && (n0 + BN <= Nn);

    for (int kt = 0; kt < K; kt += BK) {
        if (mn_full && (kt + BK <= K)) {
            // Async staging path: memory -> LDS directly, no VGPR bounce.
#pragma unroll
            for (int i = tid; i < BM * BK; i += 128) {
                int r = i >> 4, c = i & 15;
                async_g2l_b32(lds_addr_of(&As[r][c]),
                              &Ap[(long)(m0 + r) * Ars + (long)(kt + c) * Acs]);
            }
#pragma unroll
            for (int i = tid; i < BK * BN; i += 128) {
                int r = i >> 6, c = i & 63;
                async_g2l_b32(lds_addr_of(&Bs[r][c]),
                              &Bp[(long)(kt + r) * Brs + (long)(n0 + c) * Bcs]);
            }
            async_wait_all();
        } else {
            // Guarded fallback for ragged tiles.
#pragma unroll
            for (int i = tid; i < BM * BK; i += 128) {
                int r = i >> 4, c = i & 15;
                int gm = m0 + r, gk = kt + c;
                As[r][c] = (gm < M && gk < K) ? Ap[(long)gm * Ars + (long)gk * Acs] : 0.f;
            }
#pragma unroll
            for (int i = tid; i < BK * BN; i += 128) {
                int r = i >> 6, c = i & 63;
                int gk = kt + r, gn = n0 + c;
                Bs[r][c] = (gk < K && gn < Nn) ? Bp[(long)gk * Brs + (long)gn * Bcs] : 0.f;
            }
        }
        __syncthreads();

        // Prefetch next K-tile into caches (gfx1250 global_prefetch_b8)
        if (kt + BK < K) {
            int rA = tid >> 4, cA = tid & 15;
            int gm = m0 + rA, gk = kt + BK + cA;
            if (gm < M && gk < K)
                __builtin_prefetch(&Ap[(long)gm * Ars + (long)gk * Acs], 0, 1);
            int rB = tid >> 6, cB = tid & 63;
            int gk2 = kt + BK + rB, gn = n0 + cB;
            if (gk2 < K && gn < Nn)
                __builtin_prefetch(&Bp[(long)gk2 * Brs + (long)gn * Bcs], 0, 1);
        }

#pragma unroll
        for (int kk = 0; kk < BK; kk += 4) {
#pragma unroll
            for (int ii = 0; ii < 2; ++ii) {
                v2f a;
                a.x = As[wr + ii * 16 + lr][kk + 2 * half + 0];
                a.y = As[wr + ii * 16 + lr][kk + 2 * half + 1];
#pragma unroll
                for (int jj = 0; jj < 2; ++jj) {
                    v2f b;
                    b.x = Bs[kk + 2 * half + 0][wc + jj * 16 + lr];
                    b.y = Bs[kk + 2 * half + 1][wc + jj * 16 + lr];
                    acc[ii][jj] = __builtin_amdgcn_wmma_f32_16x16x4_f32(
                        false, a, false, b, (short)0, acc[ii][jj], false, false);
                }
            }
        }
        __syncthreads();
    }

    // Store (C/D layout: VGPR r -> row r (lanes 0-15) / r+8 (lanes 16-31))
#pragma unroll
    for (int ii = 0; ii < 2; ++ii)
#pragma unroll
        for (int jj = 0; jj < 2; ++jj)
#pragma unroll
            for (int r = 0; r < 8; ++r) {
                int row = m0 + wr + ii * 16 + r + 8 * half;
                int col = n0 + wc + jj * 16 + lr;
                if (row < M && col < Nn)
                    Cp[(long)row * Nn + col] = acc[ii][jj][r];
            }
}

// ---------------------------------------------------------------------------
// Implicit-GEMM 3D conv with WMMA. M = out-channels, N = output positions,
// K = Ci*27. mode==1 fuses x = concat(att*mask, enc) on-the-fly; relu flag
// fuses the activation. Weight tile staged via async-to-LDS (rows of W are
// contiguous); the im2col gather stays on the guarded scalar path.
// ---------------------------------------------------------------------------
__global__ __launch_bounds__(128)
void conv3d_wmma(const float* __restrict__ Wt,   // [Co][Ci*27] row-major
                 const float* __restrict__ Xin,  // mode 0: [N][Ci][Din^3]
                 const float* __restrict__ att,  // mode 1: [N][Ci/2][Din^3]
                 const float* __restrict__ enc,  // mode 1: [N][Ci/2][Din^3]
                 const float* __restrict__ mask, // mode 1: [N][Din^3]
                 float* __restrict__ Y,          // [N][Co][Dout^3]
                 int Co, int Ci, int Din, int Dout, int stride, int pad,
                 int mode, int relu)
{
    __shared__ float As[BM][BK + 1];
    __shared__ float Bs[BK][BN + 1];

    const int nb = blockIdx.z;
    const int K = Ci * 27;
    const int Sin = Din * Din * Din;
    const int Sout = Dout * Dout * Dout;
    const int Ch = Ci >> 1;

    const int m0 = blockIdx.y * BM;
    const int n0 = blockIdx.x * BN;
    const int tid = threadIdx.x;
    const int wave = tid >> 5;
    const int lane = tid & 31;
    const int wr = (wave >> 1) * 32;
    const int wc = (wave & 1) * 32;
    const int half = lane >> 4;
    const int lr = lane & 15;

    v8f acc[2][2];
#pragma unroll
    for (int i = 0; i < 2; ++i)
#pragma unroll
        for (int j = 0; j < 2; ++j)
            acc[i][j] = (v8f){0.f,0.f,0.f,0.f,0.f,0.f,0.f,0.f};

    for (int kt = 0; kt < K; kt += BK) {
        // Weights: contiguous rows of W [Co][K] -> async copy when full tile
        if (m0 + BM <= Co && kt + BK <= K) {
#pragma unroll
            for (int i = tid; i < BM * BK; i += 128) {
                int r = i >> 4, c = i & 15;
                async_g2l_b32(lds_addr_of(&As[r][c]),
                              &Wt[(long)(m0 + r) * K + (kt + c)]);
            }
        } else {
#pragma unroll
            for (int i = tid; i < BM * BK; i += 128) {
                int r = i >> 4, c = i & 15;
                int gm = m0 + r, gk = kt + c;
                As[r][c] = (gm < Co && gk < K) ? Wt[(long)gm * K + gk] : 0.f;
            }
        }
        // Input patch gather (im2col on the fly, concat+gate fused)
#pragma unroll
        for (int i = tid; i < BK * BN; i += 128) {
            int r = i >> 6, c = i & 63;
            int gk = kt + r, pos = n0 + c;
            float v = 0.f;
            if (gk < K && pos < Sout) {
                int ic = gk / 27, rr = gk % 27;
                int kd = rr / 9, kh = (rr / 3) % 3, kw = rr % 3;
                int od = pos / (Dout * Dout);
                int rem = pos % (Dout * Dout);
                int oh = rem / Dout, ow = rem % Dout;
                int id = od * stride - pad + kd;
                int ih = oh * stride - pad + kh;
                int iw = ow * stride - pad + kw;
                if (id >= 0 && id < Din && ih >= 0 && ih < Din && iw >= 0 && iw < Din) {
                    long ipos = ((long)id * Din + ih) * Din + iw;
                    if (mode == 0) {
                        v = Xin[((long)nb * Ci + ic) * Sin + ipos];
                    } else {
                        if (ic < Ch)
                            v = att[((long)nb * Ch + ic) * Sin + ipos] *
                                mask[(long)nb * Sin + ipos];
                        else
                            v = enc[((long)nb * Ch + (ic - Ch)) * Sin + ipos];
                    }
                }
            }
            Bs[r][c] = v;
        }
        async_wait_all();
        __syncthreads();

        if (kt + BK < K) {
            int rA = tid >> 4, cA = tid & 15;
            int gm = m0 + rA, gk = kt + BK + cA;
            if (gm < Co && gk < K)
                __builtin_prefetch(&Wt[(long)gm * K + gk], 0, 1);
        }

#pragma unroll
        for (int kk = 0; kk < BK; kk += 4) {
#pragma unroll
            for (int ii = 0; ii < 2; ++ii) {
                v2f a;
                a.x = As[wr + ii * 16 + lr][kk + 2 * half + 0];
                a.y = As[wr + ii * 16 + lr][kk + 2 * half + 1];
#pragma unroll
                for (int jj = 0; jj < 2; ++jj) {
                    v2f b;
                    b.x = Bs[kk + 2 * half + 0][wc + jj * 16 + lr];
                    b.y = Bs[kk + 2 * half + 1][wc + jj * 16 + lr];
                    acc[ii][jj] = __builtin_amdgcn_wmma_f32_16x16x4_f32(
                        false, a, false, b, (short)0, acc[ii][jj], false, false);
                }
            }
        }
        __syncthreads();
    }

#pragma unroll
    for (int ii = 0; ii < 2; ++ii)
#pragma unroll
        for (int jj = 0; jj < 2; ++jj)
#pragma unroll
            for (int r = 0; r < 8; ++r) {
                int row = m0 + wr + ii * 16 + r + 8 * half;
                int col = n0 + wc + jj * 16 + lr;
                if (row < Co && col < Sout) {
                    float v = acc[ii][jj][r];
                    if (relu) v = v > 0.f ? v : 0.f;
                    Y[((long)nb * Co + row) * Sout + col] = v;
                }
            }
}

// ---------------------------------------------------------------------------
// Softmax over axis=1 of A[N][L][L] (column softmax). One thread per (n, m).
// ---------------------------------------------------------------------------
__global__ __launch_bounds__(256)
void softmax_col(const float* __restrict__ A, float* __restrict__ O, int L, int Nb)
{
    long t = (long)blockIdx.x * blockDim.x + threadIdx.x;
    if (t >= (long)Nb * L) return;
    int n = (int)(t / L), m = (int)(t % L);
    const float* base = A + (long)n * L * L + m;
    float mx = -3.402823e38f;
    for (int l = 0; l < L; ++l) mx = fmaxf(mx, base[(long)l * L]);
    float s = 0.f;
    for (int l = 0; l < L; ++l) s += expf(base[(long)l * L] - mx);
    float inv = 1.f / s;
    float* ob = O + (long)n * L * L + m;
    for (int l = 0; l < L; ++l) ob[(long)l * L] = expf(base[(long)l * L] - mx) * inv;
}

// ---------------------------------------------------------------------------
// Row softmax of A (contiguous rows), output row-major (used transposed in
// att2 via GEMM strides). One 256-thread block per row.
// ---------------------------------------------------------------------------
__global__ __launch_bounds__(256)
void softmax_row(const float* __restrict__ A, float* __restrict__ O, int L)
{
    __shared__ float red[256];
    long row = blockIdx.x;
    const float* base = A + row * (long)L;
    float mx = -3.402823e38f;
    for (int i = threadIdx.x; i < L; i += 256) mx = fmaxf(mx, base[i]);
    red[threadIdx.x] = mx; __syncthreads();
    for (int s = 128; s > 0; s >>= 1) {
        if ((int)threadIdx.x < s) red[threadIdx.x] = fmaxf(red[threadIdx.x], red[threadIdx.x + s]);
        __syncthreads();
    }
    mx = red[0]; __syncthreads();
    float sum = 0.f;
    for (int i = threadIdx.x; i < L; i += 256) sum += expf(base[i] - mx);
    red[threadIdx.x] = sum; __syncthreads();
    for (int s = 128; s > 0; s >>= 1) {
        if ((int)threadIdx.x < s) red[threadIdx.x] += red[threadIdx.x + s];
        __syncthreads();
    }
    float inv = 1.f / red[0];
    float* ob = O + row * (long)L;
    for (int i = threadIdx.x; i < L; i += 256) ob[i] = expf(base[i] - mx) * inv;
}

// mask[n][p] = sigmoid(sum_c att[n][c][p] * wg[c])  (1x1x1 gate conv)
__global__ __launch_bounds__(256)
void gate_mask(const float* __restrict__ att, const float* __restrict__ wg,
               float* __restrict__ mask, int C, int L, int Nb)
{
    long t = (long)blockIdx.x * blockDim.x + threadIdx.x;
    if (t >= (long)Nb * L) return;
    int n = (int)(t / L), p = (int)(t % L);
    float s = 0.f;
    for (int c = 0; c < C; ++c) s += att[((long)n * C + c) * L + p] * wg[c];
    mask[t] = 1.f / (1.f + expf(-s));
}

// Per-channel batch statistics over (N, spatial)
__global__ __launch_bounds__(256)
void bn_stats(const float* __restrict__ Y, float* __restrict__ mean,
              float* __restrict__ var, int C, int S, int Nb)
{
    __shared__ float ssum[256];
    __shared__ float ssq[256];
    int c = blockIdx.x;
    int total = Nb * S;
    float s = 0.f, q = 0.f;
    for (int i = threadIdx.x; i < total; i += 256) {
        int n = i / S, p = i % S;
        float v = Y[((long)n * C + c) * S + p];
        s += v; q += v * v;
    }
    ssum[threadIdx.x] = s; ssq[threadIdx.x] = q; __syncthreads();
    for (int st = 128; st > 0; st >>= 1) {
        if ((int)threadIdx.x < st) {
            ssum[threadIdx.x] += ssum[threadIdx.x + st];
            ssq[threadIdx.x]  += ssq[threadIdx.x + st];
        }
        __syncthreads();
    }
    if (threadIdx.x == 0) {
        float m = ssum[0] / (float)total;
        mean[c] = m;
        var[c] = ssq[0] / (float)total - m * m;
    }
}

__global__ __launch_bounds__(256)
void bn_relu(float* __restrict__ Y, const float* __restrict__ mean,
             const float* __restrict__ var, const float* __restrict__ g,
             const float* __restrict__ b, int C, int S, long total)
{
    long i = (long)blockIdx.x * blockDim.x + threadIdx.x;
    if (i >= total) return;
    int c = (int)((i / S) % C);
    float v = (Y[i] - mean[c]) * rsqrtf(var[c] + 1e-5f) * g[c] + b[c];
    Y[i] = v > 0.f ? v : 0.f;
}

// Final heads: flat = concat(z1,z2,axis=1).reshape(-1, 768) (torch semantics),
// out[0:432] = sigmoid(flat@Wcls^T), out[432:] = sigmoid(flat@Wsurv^T)
__global__ __launch_bounds__(256)
void head(const float* __restrict__ z1, const float* __restrict__ z2,
          const float* __restrict__ Wcls, const float* __restrict__ Wsurv,
          float* __restrict__ out)
{
    const int ROWS = 432, CC = 768, S = 216, HC = 384;
    const long PB = (long)CC * S;   // per-batch flat block = 165888
    int t = blockIdx.x * blockDim.x + threadIdx.x;
    if (t >= ROWS * 11) return;
    int r = t / 11, o = t % 11;
    const float* w = (o == 0) ? Wcls : (Wsurv + (long)(o - 1) * CC);
    float s = 0.f;
    long base = (long)r * CC;
    for (int j = 0; j < CC; ++j) {
        long i = base + j;
        int n = (int)(i / PB);
        long rem = i % PB;
        int c = (int)(rem / S), sp = (int)(rem % S);
        float v = (c < HC) ? z1[((long)n * HC + c) * S + sp]
                           : z2[((long)n * HC + (c - HC)) * S + sp];
        s += v * w[j];
    }
    float sig = 1.f / (1.f + expf(-s));
    if (o == 0) out[r] = sig;
    else out[432 + (long)r * 10 + (o - 1)] = sig;
}

// ---------------------------------------------------------------------------
extern "C" void kernel_launch(void* const* d_in, const int* in_sizes, int n_in,
                              void* d_out, int out_size, void* d_ws, size_t ws_size,
                              hipStream_t stream)
{
    const int Nb = 2, C = 768, D = 16;
    const int L = D * D * D;             // 4096
    (void)in_sizes; (void)n_in; (void)out_size; (void)ws_size;

    const float* enc1  = (const float*)d_in[0];
    const float* enc2  = (const float*)d_in[1];
    const float* W_e   = (const float*)d_in[2];
    const float* W_g   = (const float*)d_in[3];
    const float* W1    = (const float*)d_in[4];
    const float* W2    = (const float*)d_in[5];
    const float* W3    = (const float*)d_in[6];
    const float* W4    = (const float*)d_in[7];
    const float* bn1w  = (const float*)d_in[8];
    const float* bn1b  = (const float*)d_in[9];
    const float* bn2w  = (const float*)d_in[10];
    const float* bn2b  = (const float*)d_in[11];
    const float* Wcls  = (const float*)d_in[12];
    const float* Wsurv = (const float*)d_in[13];
    float* out = (float*)d_out;

    // Workspace carve-up (floats)
    float* ws = (float*)d_ws;
    size_t off = 0;
    float* corr = ws + off;  off += (size_t)Nb * L * C;        // [N][L][C]
    float* Amat = ws + off;  off += (size_t)Nb * L * L;        // [N][L][L]
    float* A1   = ws + off;  off += (size_t)Nb * L * L;        // col-softmax
    float* Brow = ws + off;  off += (size_t)Nb * L * L;        // row-softmax (row-major)
    float* att1 = ws + off;  off += (size_t)Nb * C * L;
    float* att2 = ws + off;  off += (size_t)Nb * C * L;
    float* m1   = ws + off;  off += (size_t)Nb * L;
    float* m2   = ws + off;  off += (size_t)Nb * L;
    float* y1   = ws + off;  off += (size_t)Nb * C * 512;      // conv1 out 8^3
    float* y2   = ws + off;  off += (size_t)Nb * C * 512;
    float* mu1  = ws + off;  off += C;
    float* vr1  = ws + off;  off += C;
    float* mu2  = ws + off;  off += C;
    float* vr2  = ws + off;  off += C;
    float* z1   = ws + off;  off += (size_t)Nb * (C/2) * 216;  // conv3 out 6^3
    float* z2   = ws + off;  off += (size_t)Nb * (C/2) * 216;

    // 1) corr[n][l][k] = sum_c f2[n][c][l] * W_e[k][c]
    //    A-op = F2^T (Ars=1, Acs=L), B-op = W_e^T (Brs=1, Bcs=C)
    gemm_f32_wmma<<<dim3(C/BN, L/BM, Nb), 128, 0, stream>>>(
        enc2, W_e, corr, L, C, C,
        1L, (long)L, 1L, (long)C,
        (long)C * L, 0L, (long)L * C);

    // 2) A[n][l][m] = sum_c corr[n][l][c] * f1[n][c][m]
    gemm_f32_wmma<<<dim3(L/BN, L/BM, Nb), 128, 0, stream>>>(
        corr, enc1, Amat, L, L, C,
        (long)C, 1L, (long)L, 1L,
        (long)L * C, (long)C * L, (long)L * L);

    // 3) softmaxes
    softmax_col<<<(Nb * L + 255) / 256, 256, 0, stream>>>(Amat, A1, L, Nb);
    softmax_row<<<Nb * L, 256, 0, stream>>>(Amat, Brow, L);

    // 4) att1[n][c][m] = sum_l f2[n][c][l] * A1[n][l][m]
    gemm_f32_wmma<<<dim3(L/BN, C/BM, Nb), 128, 0, stream>>>(
        enc2, A1, att1, C, L, L,
        (long)L, 1L, (long)L, 1L,
        (long)C * L, (long)L * L, (long)C * L);

    // 5) att2[n][c][m] = sum_l f1[n][c][l] * Brow[n][m][l]  (B transposed)
    gemm_f32_wmma<<<dim3(L/BN, C/BM, Nb), 128, 0, stream>>>(
        enc1, Brow, att2, C, L, L,
        (long)L, 1L, 1L, (long)L,
        (long)C * L, (long)L * L, (long)C * L);

    // 6) gate masks
    gate_mask<<<(Nb * L + 255) / 256, 256, 0, stream>>>(att1, W_g, m1, C, L, Nb);
    gate_mask<<<(Nb * L + 255) / 256, 256, 0, stream>>>(att2, W_g, m2, C, L, Nb);

    // 7) conv1/conv2: 1536->768, 3^3, stride 2, pad 1 (concat+gate fused)
    conv3d_wmma<<<dim3((512 + BN - 1)/BN, C/BM, Nb), 128, 0, stream>>>(
        W1, nullptr, att1, enc1, m1, y1, C, 2*C, 16, 8, 2, 1, 1, 0);
    conv3d_wmma<<<dim3((512 + BN - 1)/BN, C/BM, Nb), 128, 0, stream>>>(
        W2, nullptr, att2, enc2, m2, y2, C, 2*C, 16, 8, 2, 1, 1, 0);

    // 8) training-mode BN + ReLU
    bn_stats<<<C, 256, 0, stream>>>(y1, mu1, vr1, C, 512, Nb);
    bn_stats<<<C, 256, 0, stream>>>(y2, mu2, vr2, C, 512, Nb);
    long tot1 = (long)Nb * C * 512;
    bn_relu<<<(int)((tot1 + 255) / 256), 256, 0, stream>>>(y1, mu1, vr1, bn1w, bn1b, C, 512, tot1);
    bn_relu<<<(int)((tot1 + 255) / 256), 256, 0, stream>>>(y2, mu2, vr2, bn2w, bn2b, C, 512, tot1);

    // 9) conv3/conv4: 768->384, 3^3, stride 1, pad 0, ReLU fused
    conv3d_wmma<<<dim3((216 + BN - 1)/BN, (C/2)/BM, Nb), 128, 0, stream>>>(
        W3, y1, nullptr, nullptr, nullptr, z1, C/2, C, 8, 6, 1, 0, 0, 1);
    conv3d_wmma<<<dim3((216 + BN - 1)/BN, (C/2)/BM, Nb), 128, 0, stream>>>(
        W4, y2, nullptr, nullptr, nullptr, z2, C/2, C, 8, 6, 1, 0, 0, 1);

    // 10) sigmoid heads
    head<<<(432 * 11 + 255) / 256, 256, 0, stream>>>(z1, z2, Wcls, Wsurv, out);
}